// MyGraphAttentionLayer_35794257445172
// MI455X (gfx1250) — compile-verified
//
#include <hip/hip_runtime.h>

#define N_NODES 12288
#define IN_F    512
#define OUT_F   64
#define ALPHA   0.2f

typedef __attribute__((ext_vector_type(2))) float v2f;
typedef __attribute__((ext_vector_type(8))) float v8f;

// ---------------------------------------------------------------------------
// Kernel 1: h = x @ W  via V_WMMA_F32_16X16X4_F32 (fp32-exact matrix pipe).
// One wave computes a 16x64 tile of h. 8 waves / block -> 128 rows / block.
// ---------------------------------------------------------------------------
__global__ __launch_bounds__(256) void gat_gemm_wmma(const float* __restrict__ x,
                                                     const float* __restrict__ W,
                                                     float* __restrict__ h) {
  const int lane = threadIdx.x & 31;
  const int wave = threadIdx.x >> 5;
  const int kh   = lane >> 4;   // lane half selects K+0/1 vs K+2/3
  const int l16  = lane & 15;
  const int row0 = (blockIdx.x * 8 + wave) * 16;

  v8f acc0 = {}, acc1 = {}, acc2 = {}, acc3 = {};
  const float* xrow = x + (size_t)(row0 + l16) * IN_F;

  for (int k = 0; k < IN_F; k += 4) {
    v2f a = *(const v2f*)(xrow + k + 2 * kh);                 // x[m][k'..k'+1]
    const float* wk = W + (size_t)(k + 2 * kh) * OUT_F + l16; // W[k'][n], W[k'+1][n]
    v2f b0 = { wk[ 0], wk[OUT_F +  0] };
    v2f b1 = { wk[16], wk[OUT_F + 16] };
    v2f b2 = { wk[32], wk[OUT_F + 32] };
    v2f b3 = { wk[48], wk[OUT_F + 48] };
    acc0 = __builtin_amdgcn_wmma_f32_16x16x4_f32(false, a, false, b0, (short)0, acc0, false, false);
    acc1 = __builtin_amdgcn_wmma_f32_16x16x4_f32(false, a, false, b1, (short)0, acc1, false, false);
    acc2 = __builtin_amdgcn_wmma_f32_16x16x4_f32(false, a, false, b2, (short)0, acc2, false, false);
    acc3 = __builtin_amdgcn_wmma_f32_16x16x4_f32(false, a, false, b3, (short)0, acc3, false, false);
  }

  float* hp = h + (size_t)(row0 + 8 * kh) * OUT_F + l16;
#pragma unroll
  for (int r = 0; r < 8; ++r) {
    hp[r * OUT_F +  0] = acc0[r];
    hp[r * OUT_F + 16] = acc1[r];
    hp[r * OUT_F + 32] = acc2[r];
    hp[r * OUT_F + 48] = acc3[r];
  }
}

// ---------------------------------------------------------------------------
// Kernel 2: s_nei[i] = h[i,:] . a[0:64],  s_self[i] = h[i,:] . a[64:128]
// ---------------------------------------------------------------------------
__global__ void gat_scores(const float* __restrict__ h, const float* __restrict__ a,
                           float* __restrict__ s_nei, float* __restrict__ s_self) {
  int i = blockIdx.x * blockDim.x + threadIdx.x;
  if (i >= N_NODES) return;
  const float* hr = h + (size_t)i * OUT_F;
  float sn = 0.f, ss = 0.f;
#pragma unroll 8
  for (int f = 0; f < OUT_F; ++f) {
    float v = hr[f];
    sn = fmaf(v, a[f], sn);
    ss = fmaf(v, a[OUT_F + f], ss);
  }
  s_nei[i]  = sn;
  s_self[i] = ss;
}

// bits [3:0] = nonzero flags of the 4 bytes of u (adj bytes are 0/1 bools)
__device__ __forceinline__ unsigned int bytemask4(unsigned int u) {
  return ((u & 0x01010101u) * 0x01020408u) >> 24;
}

// ---------------------------------------------------------------------------
// Kernel 3: masked softmax + aggregation. One wave32 per row.
// Pass 1 (HBM stream, 144 MB): row max over adj hits; each lane loads 16 adj
//   bytes per iter -> 512 B coalesced per wave, 24 iters/row.
// Pass 2 (L2-resident — 144 MB adj fits in 192 MB L2): only the 16-bit hit
//   mask is broadcast; every lane recomputes the scalar weight from the
//   uniform-address s_nei[j] load, then 32 lanes cooperatively accumulate the
//   64-wide feature row (coalesced 256 B h[j] gathers, h is L2-resident).
// ---------------------------------------------------------------------------
__global__ __launch_bounds__(256) void gat_attend(const unsigned char* __restrict__ adj,
                                                  const float* __restrict__ h,
                                                  const float* __restrict__ s_nei,
                                                  const float* __restrict__ s_self,
                                                  float* __restrict__ out) {
  const int lane = threadIdx.x & 31;
  const int wave = threadIdx.x >> 5;
  const int row  = blockIdx.x * 8 + wave;
  const unsigned char* arow = adj + (size_t)row * N_NODES;
  const float ss = s_self[row];

  // ---- pass 1: row max of leaky_relu scores ----
  float m = -1e30f;
  for (int base = 0; base < N_NODES; base += 512) {
    __builtin_prefetch(arow + base + 4096, 0, 0);     // global_prefetch_b8, far ahead
    uint4 q = *(const uint4*)(arow + base + lane * 16);
    if (q.x | q.y | q.z | q.w) {
      unsigned int hm = bytemask4(q.x) | (bytemask4(q.y) << 4) |
                        (bytemask4(q.z) << 8) | (bytemask4(q.w) << 12);
      while (hm) {
        int c = __builtin_ctz(hm);
        hm &= hm - 1;
        float e = ss + s_nei[base + lane * 16 + c];
        e = e > 0.f ? e : ALPHA * e;
        m = fmaxf(m, e);
      }
    }
  }
#pragma unroll
  for (int off = 16; off >= 1; off >>= 1)
    m = fmaxf(m, __shfl_xor(m, off, 32));

  // ---- pass 2: exp-weights + cooperative aggregation ----
  float S = 0.f, acc0 = 0.f, acc1 = 0.f;    // S kept uniform across all lanes
  for (int base = 0; base < N_NODES; base += 512) {
    uint4 q = *(const uint4*)(arow + base + lane * 16);
    unsigned int hm = 0;
    if (q.x | q.y | q.z | q.w)
      hm = bytemask4(q.x) | (bytemask4(q.y) << 4) |
           (bytemask4(q.z) << 8) | (bytemask4(q.w) << 12);

    unsigned int ballot = __builtin_amdgcn_ballot_w32(hm != 0);
    while (ballot) {                          // uniform loop: all lanes share ballot
      int src = __builtin_ctz(ballot);
      ballot &= ballot - 1;
      unsigned int shm = (unsigned int)__shfl((int)hm, src, 32);  // 16-bit hit mask
      while (shm) {                           // uniform: shm identical on all lanes
        int c = __builtin_ctz(shm);
        shm &= shm - 1;
        int j = base + src * 16 + c;
        float e = ss + s_nei[j];              // uniform-address broadcast load
        e = e > 0.f ? e : ALPHA * e;
        float pj = __expf(e - m);             // recomputed identically on all lanes
        const float* hj = h + (size_t)j * OUT_F;
        acc0 = fmaf(pj, hj[lane],      acc0); // coalesced 256 B row gather
        acc1 = fmaf(pj, hj[lane + 32], acc1);
        S += pj;
      }
    }
  }

  const float rinv = 1.0f / S;               // self-loop guarantees S > 0
  float v0 = acc0 * rinv, v1 = acc1 * rinv;
  v0 = v0 > 0.f ? v0 : __expf(v0) - 1.f;     // ELU (alpha = 1)
  v1 = v1 > 0.f ? v1 : __expf(v1) - 1.f;
  out[(size_t)row * OUT_F + lane]      = v0;
  out[(size_t)row * OUT_F + lane + 32] = v1;
}

// ---------------------------------------------------------------------------
extern "C" void kernel_launch(void* const* d_in, const int* in_sizes, int n_in,
                              void* d_out, int out_size, void* d_ws, size_t ws_size,
                              hipStream_t stream) {
  const float*         x   = (const float*)d_in[0];
  const unsigned char* adj = (const unsigned char*)d_in[1];   // jax bool = 1 byte
  const float*         W   = (const float*)d_in[2];
  const float*         a   = (const float*)d_in[3];
  float*               out = (float*)d_out;

  float* h      = (float*)d_ws;                    // 12288*64 fp32 = 3 MB
  float* s_nei  = h + (size_t)N_NODES * OUT_F;     // 48 KB
  float* s_self = s_nei + N_NODES;                 // 48 KB

  gat_gemm_wmma<<<N_NODES / 128, 256, 0, stream>>>(x, W, h);
  gat_scores  <<<(N_NODES + 255) / 256, 256, 0, stream>>>(h, a, s_nei, s_self);
  gat_attend  <<<N_NODES / 8, 256, 0, stream>>>(adj, h, s_nei, s_self, out);
}